// CenterLoss_with_delta_37666863186512
// MI455X (gfx1250) — compile-verified
//
#include <hip/hip_runtime.h>
#include <stdint.h>
#include <math.h>

typedef __attribute__((ext_vector_type(2))) float v2f;
typedef __attribute__((ext_vector_type(8))) float v8f;

#define TPB 256
#define FEAT_D 2048
#define EPT 8  // FEAT_D / TPB elements per thread

__device__ __forceinline__ void atomic_add_f32(float* p, float v) {
    // relaxed + agent scope, result discarded -> no-return global_atomic_add_f32
    __hip_atomic_fetch_add(p, v, __ATOMIC_RELAXED, __HIP_MEMORY_SCOPE_AGENT);
}

// CDNA5 async copy engine: 16B per lane, global -> LDS, tracked by ASYNCcnt.
// GVS addressing: mem = SADDR(64b SGPR) + VADDR(32b VGPR byte offset).
__device__ __forceinline__ void async_copy_b128(uint32_t lds_byte_addr,
                                                uint32_t glb_byte_off,
                                                const void* base) {
    asm volatile("global_load_async_to_lds_b128 %0, %1, %2"
                 :: "v"(lds_byte_addr), "v"(glb_byte_off), "s"(base)
                 : "memory");
}

__device__ __forceinline__ void wait_asynccnt_le2() {
    asm volatile("s_wait_asynccnt 2" ::: "memory");
}
__device__ __forceinline__ void wait_asynccnt_0() {
    asm volatile("s_wait_asynccnt 0" ::: "memory");
}

__global__ void __launch_bounds__(TPB) zero_kernel(float* a, long long na,
                                                   float* b, long long nb) {
    long long i = (long long)blockIdx.x * blockDim.x + threadIdx.x;
    long long stride = (long long)gridDim.x * blockDim.x;
    for (long long k = i; k < na; k += stride) a[k] = 0.0f;
    for (long long k = i; k < nb; k += stride) b[k] = 0.0f;
}

// One block processes rows b = blockIdx.x, blockIdx.x+gridDim.x, ...
// Each row (2048 f32 = 8 KB) is streamed HBM->LDS by the async copy engine,
// double-buffered: issue row r+stride, then s_wait_asynccnt 2 so row r is ready.
__global__ void __launch_bounds__(TPB) center_main_kernel(
    const int* __restrict__ y, const float* __restrict__ feat,
    const float* __restrict__ centers, float* __restrict__ sums,
    float* __restrict__ counts, float* __restrict__ sumsq, int batch)
{
    __shared__ float buf[2][FEAT_D];
    __shared__ float part[TPB];

    const int t = threadIdx.x;
    const int stride = gridDim.x;
    const int col0 = t * EPT;

    const uint32_t ldsA = (uint32_t)(uintptr_t)&buf[0][col0];
    const uint32_t ldsB = (uint32_t)(uintptr_t)&buf[1][col0];

    int row = blockIdx.x;
    if (row < batch) {
        uint32_t off = (uint32_t)(row * FEAT_D + col0) * 4u;
        async_copy_b128(ldsA, off, feat);
        async_copy_b128(ldsA + 16u, off + 16u, feat);
    }

    float acc = 0.0f;
    int cur = 0;
    for (; row < batch; row += stride) {
        const int nxt = row + stride;
        if (nxt < batch) {
            uint32_t lds = (cur == 0) ? ldsB : ldsA;
            uint32_t off = (uint32_t)(nxt * FEAT_D + col0) * 4u;
            async_copy_b128(lds, off, feat);
            async_copy_b128(lds + 16u, off + 16u, feat);
            wait_asynccnt_le2();   // 2 newest outstanding -> current buffer done
        } else {
            wait_asynccnt_0();
        }

        const int c = y[row];
        if (t == 0) atomic_add_f32(&counts[c], 1.0f);

        const float* __restrict__ crow = centers + (size_t)c * FEAT_D;
        float*       __restrict__ srow = sums    + (size_t)c * FEAT_D;
        const float* bp = &buf[cur][col0];
#pragma unroll
        for (int k = 0; k < EPT; ++k) {
            float f = bp[k];
            float d = f - crow[col0 + k];
            acc += d * d;                       // loss partial
            atomic_add_f32(&srow[col0 + k], f); // segment-sum scatter (L2 RMW)
        }
        cur ^= 1;
    }

    // Block reduction of 256 loss partials: fold with v_wmma_f32_16x16x4_f32.
    // A = all-ones 16x4 -> D[i][n] = sum_k B[k][n]; every D row identical, so
    // sum(c[0] over 32 lanes) = 2 * total, independent of B lane layout.
    part[t] = acc;
    __syncthreads();
    if (t < 32) {                 // wave 0: EXEC all ones (WMMA requirement)
        v2f a; a.x = 1.0f; a.y = 1.0f;
        v8f cacc = {};
#pragma unroll
        for (int j = 0; j < 4; ++j) {
            v2f b;
            b.x = part[j * 64 + t];
            b.y = part[j * 64 + 32 + t];
            cacc = __builtin_amdgcn_wmma_f32_16x16x4_f32(
                false, a, false, b, (short)0, cacc, false, false);
        }
        float s = cacc[0];
#pragma unroll
        for (int m = 16; m >= 1; m >>= 1) s += __shfl_xor(s, m, 32);
        if (t == 0) atomic_add_f32(sumsq, 0.5f * s);  // s == 2 * block total
    }
}

// In-place: d_out[1..] currently holds segment sums; convert to centers_grad.
// counts==0 -> coeff==0 -> grad row is 0 (rand_fill in reference is irrelevant).
__global__ void __launch_bounds__(TPB) finalize_kernel(
    const float* __restrict__ centers, const float* __restrict__ counts,
    const float* __restrict__ sumsq, float* __restrict__ out, int C, int batch)
{
    long long gid = (long long)blockIdx.x * blockDim.x + threadIdx.x;
    if (gid == 0) out[0] = 0.5f * sqrtf(sumsq[0]) / (float)batch;
    float* grad = out + 1;
    const long long total = (long long)C * FEAT_D;
    const long long stride = (long long)gridDim.x * blockDim.x;
    for (long long k = gid; k < total; k += stride) {
        int c = (int)(k / FEAT_D);
        float cnt = counts[c];
        float coeff = cnt / (1.0f + cnt);
        float mean = grad[k] / fmaxf(cnt, 1.0f);
        grad[k] = coeff * (centers[k] - mean);
    }
}

extern "C" void kernel_launch(void* const* d_in, const int* in_sizes, int n_in,
                              void* d_out, int out_size, void* d_ws, size_t ws_size,
                              hipStream_t stream) {
    const int*   y       = (const int*)d_in[0];
    const float* feat    = (const float*)d_in[1];
    const float* centers = (const float*)d_in[2];

    const int batch = in_sizes[0];                    // 8192
    const int D     = FEAT_D;                         // 2048 (fixed by reference)
    const int C     = in_sizes[2] / D;                // 1000

    float* out    = (float*)d_out;
    float* sums   = out + 1;            // accumulate segment sums in-place
    float* counts = (float*)d_ws;       // C floats
    float* sumsq  = counts + C;         // 1 float

    zero_kernel<<<2048, TPB, 0, stream>>>(out, 1LL + (long long)C * D,
                                          counts, (long long)C + 1);

    int grid = batch < 1024 ? batch : 1024;
    center_main_kernel<<<grid, TPB, 0, stream>>>(y, feat, centers, sums,
                                                 counts, sumsq, batch);

    finalize_kernel<<<2048, TPB, 0, stream>>>(centers, counts, sumsq, out, C, batch);
}